// GraphCNN_87187836109058
// MI455X (gfx1250) — compile-verified
//
#include <hip/hip_runtime.h>
#include <math.h>

// ---------------------------------------------------------------------------
// GCN (2-layer, GCNConv + LeakyReLU + mean-pool + softmax) for MI455X/gfx1250
//
// Sizes (fixed by the reference):
//   N_NODES=100000, N_EDGES=1600000, N_GRAPHS=64, D_IN=D_HID=128, D_OUT=4
//
// Strategy:
//  * GEMM1 (X@W1, fp32) via V_WMMA_F32_16X16X4_F32, 1 wave per 16x16 tile.
//  * Edge message passing with global f32 atomics: the working set
//    (X + H + agg1 ~ 154MB) fits in the 192MB L2, so random gather/scatter
//    stays on-chip.
//  * Layer 2 projected to D_OUT=4 BEFORE aggregation (linearity), shrinking
//    messages from 512B to 16B per edge.
// ---------------------------------------------------------------------------

#define N_NODES   100000
#define N_EDGES   1600000
#define N_GRAPHS  64
#define D_IN      128
#define D_HID     128
#define D_OUT     4
#define LEAKY     0.01f

typedef __attribute__((ext_vector_type(2))) float v2f;
typedef __attribute__((ext_vector_type(8))) float v8f;

// ------------------------------- utility -----------------------------------
__global__ void fill_kernel(float* __restrict__ p, float v, int n) {
  int i = blockIdx.x * blockDim.x + threadIdx.x;
  if (i < n) p[i] = v;
}

// deg[dst] += 1 for each real edge (deg pre-initialized to 1.0 = self-loop)
__global__ void deg_kernel(const int* __restrict__ dst, float* __restrict__ deg) {
  int e = blockIdx.x * blockDim.x + threadIdx.x;
  if (e < N_EDGES) atomicAdd(&deg[dst[e]], 1.0f);
}

__global__ void rsqrt_kernel(const float* __restrict__ deg, float* __restrict__ dis) {
  int i = blockIdx.x * blockDim.x + threadIdx.x;
  if (i < N_NODES) {
    float d = deg[i];
    dis[i] = (d > 0.0f) ? rsqrtf(d) : 0.0f;
  }
}

// --------------------------- GEMM1: H = X @ W1 ------------------------------
// Block = 256 threads = 8 wave32. blockIdx.x selects a 16-row M tile,
// wave id selects one of 8 N tiles (N = 128). K loop: 128 in steps of 4.
// A (16x4 f32): lanes 0-15 hold rows M=0..15 with K={0,1}; lanes 16-31 K={2,3}.
// B (4x16 f32): mirrored layout with N on lanes.
// C/D (16x16 f32): VGPR r -> M=r (lanes 0-15) / M=8+r (lanes 16-31), N=lane&15.
__global__ void __launch_bounds__(256)
gemm1_wmma(const float* __restrict__ X, const float* __restrict__ W1,
           float* __restrict__ H) {
  const int wave = threadIdx.x >> 5;      // 0..7 -> N tile
  const int lane = threadIdx.x & 31;
  const int half = lane >> 4;             // 0 or 1 (K split)
  const int l    = lane & 15;
  const int m0   = blockIdx.x * 16;
  const int n0   = wave * 16;

  const float* __restrict__ arow = X + (size_t)(m0 + l) * D_IN;

  v8f acc = {};
  for (int k = 0; k < D_IN; k += 4) {
    const int ka = k + half * 2;
    v2f a, b;
    a.x = arow[ka];
    a.y = arow[ka + 1];
    b.x = W1[(size_t)ka       * D_HID + n0 + l];
    b.y = W1[(size_t)(ka + 1) * D_HID + n0 + l];
    acc = __builtin_amdgcn_wmma_f32_16x16x4_f32(
        /*neg_a=*/false, a, /*neg_b=*/false, b,
        /*c_mod=*/(short)0, acc, /*reuse_a=*/false, /*reuse_b=*/false);
  }

#pragma unroll
  for (int r = 0; r < 8; ++r) {
    const int row = m0 + r + half * 8;
    H[(size_t)row * D_HID + n0 + l] = acc[r];
  }
}

// ----------------- Layer-1 aggregation: agg1[d] += w * H[s] -----------------
// One wave per edge; each lane moves a float4 (32 lanes * 16B = 512B row).
// Virtual edges [N_EDGES, N_EDGES+N_NODES) are the self-loops.
__global__ void __launch_bounds__(256)
edge_agg1(const int* __restrict__ src, const int* __restrict__ dst,
          const float* __restrict__ dis, const float* __restrict__ H,
          float* __restrict__ agg1) {
  const int e    = blockIdx.x * (blockDim.x >> 5) + (threadIdx.x >> 5);
  const int lane = threadIdx.x & 31;
  if (e >= N_EDGES + N_NODES) return;

  int s, d;
  if (e < N_EDGES) { s = src[e]; d = dst[e]; }
  else             { s = d = e - N_EDGES;    }

  const float w = dis[s] * dis[d];
  const float4 v = ((const float4*)(H + (size_t)s * D_HID))[lane];
  float* o = agg1 + (size_t)d * D_HID + lane * 4;
  atomicAdd(o + 0, w * v.x);
  atomicAdd(o + 1, w * v.y);
  atomicAdd(o + 2, w * v.z);
  atomicAdd(o + 3, w * v.w);
}

// --------- h2 = LeakyReLU(agg1 + b1) @ W2   (one thread per node) ----------
__global__ void h2_kernel(const float* __restrict__ agg1,
                          const float* __restrict__ b1,
                          const float* __restrict__ W2,
                          float* __restrict__ h2) {
  const int i = blockIdx.x * blockDim.x + threadIdx.x;
  if (i >= N_NODES) return;
  const float* __restrict__ row = agg1 + (size_t)i * D_HID;
  float a0 = 0.f, a1 = 0.f, a2 = 0.f, a3 = 0.f;
#pragma unroll 4
  for (int k = 0; k < D_HID; ++k) {
    float v = row[k] + b1[k];
    v = (v >= 0.0f) ? v : LEAKY * v;
    const float4 w = ((const float4*)W2)[k];   // W2 row k (D_OUT=4)
    a0 += v * w.x; a1 += v * w.y; a2 += v * w.z; a3 += v * w.w;
  }
  float4* o = (float4*)(h2 + (size_t)i * D_OUT);
  *o = make_float4(a0, a1, a2, a3);
}

// ----------------- Layer-2 aggregation: agg2[d] += w * h2[s] ----------------
__global__ void edge_agg2(const int* __restrict__ src, const int* __restrict__ dst,
                          const float* __restrict__ dis,
                          const float* __restrict__ h2,
                          float* __restrict__ agg2) {
  const int e = blockIdx.x * blockDim.x + threadIdx.x;
  if (e >= N_EDGES + N_NODES) return;
  int s, d;
  if (e < N_EDGES) { s = src[e]; d = dst[e]; }
  else             { s = d = e - N_EDGES;    }
  const float w = dis[s] * dis[d];
  const float4 v = *(const float4*)(h2 + (size_t)s * D_OUT);
  float* o = agg2 + (size_t)d * D_OUT;
  atomicAdd(o + 0, w * v.x);
  atomicAdd(o + 1, w * v.y);
  atomicAdd(o + 2, w * v.z);
  atomicAdd(o + 3, w * v.w);
}

// -------------------------- mean pool (sums) --------------------------------
__global__ void pool_kernel(const int* __restrict__ batch,
                            const float* __restrict__ agg2,
                            float* __restrict__ pooled,
                            float* __restrict__ cnt) {
  const int i = blockIdx.x * blockDim.x + threadIdx.x;
  if (i >= N_NODES) return;
  const int g = batch[i];
  const float4 v = *(const float4*)(agg2 + (size_t)i * D_OUT);
  atomicAdd(&pooled[g * D_OUT + 0], v.x);
  atomicAdd(&pooled[g * D_OUT + 1], v.y);
  atomicAdd(&pooled[g * D_OUT + 2], v.z);
  atomicAdd(&pooled[g * D_OUT + 3], v.w);
  atomicAdd(&cnt[g], 1.0f);
}

// ----------------- mean + bias + softmax -> d_out (64x4) --------------------
__global__ void final_kernel(const float* __restrict__ pooled,
                             const float* __restrict__ cnt,
                             const float* __restrict__ b2,
                             float* __restrict__ out) {
  const int g = blockIdx.x * blockDim.x + threadIdx.x;
  if (g >= N_GRAPHS) return;
  const float c = fmaxf(cnt[g], 1.0f);
  float v[D_OUT];
  float m = -3.402823466e+38f;
#pragma unroll
  for (int f = 0; f < D_OUT; ++f) {
    v[f] = pooled[g * D_OUT + f] / c + b2[f];
    m = fmaxf(m, v[f]);
  }
  float s = 0.0f;
#pragma unroll
  for (int f = 0; f < D_OUT; ++f) { v[f] = __expf(v[f] - m); s += v[f]; }
#pragma unroll
  for (int f = 0; f < D_OUT; ++f) out[g * D_OUT + f] = v[f] / s;
}

// ---------------------------------------------------------------------------
extern "C" void kernel_launch(void* const* d_in, const int* in_sizes, int n_in,
                              void* d_out, int out_size, void* d_ws, size_t ws_size,
                              hipStream_t stream) {
  (void)in_sizes; (void)n_in; (void)out_size; (void)ws_size;

  const float* x    = (const float*)d_in[0];
  const int*   ei   = (const int*)  d_in[1];   // (2, N_EDGES) row-major
  const int*   bat  = (const int*)  d_in[2];
  const float* W1   = (const float*)d_in[3];
  const float* b1   = (const float*)d_in[4];
  const float* W2   = (const float*)d_in[5];
  const float* b2   = (const float*)d_in[6];
  float*       out  = (float*)d_out;

  const int* src = ei;            // edge_index[0]
  const int* dst = ei + N_EDGES;  // edge_index[1]

  // Workspace layout (floats)
  float* ws     = (float*)d_ws;
  float* H      = ws;                                   // 12,800,000
  float* agg1   = H      + (size_t)N_NODES * D_HID;     // 12,800,000
  float* deg    = agg1   + (size_t)N_NODES * D_HID;     //    100,000
  float* dis    = deg    + N_NODES;                     //    100,000
  float* h2     = dis    + N_NODES;                     //    400,000
  float* agg2   = h2     + (size_t)N_NODES * D_OUT;     //    400,000
  float* pooled = agg2   + (size_t)N_NODES * D_OUT;     //        256
  float* cnt    = pooled + N_GRAPHS * D_OUT;            //         64

  const int T = 256;
  const int nAgg1 = N_NODES * D_HID;

  // 1. (re)initialize accumulators every call — required for determinism
  fill_kernel<<<(nAgg1 + T - 1) / T, T, 0, stream>>>(agg1, 0.0f, nAgg1);
  fill_kernel<<<(N_NODES * D_OUT + T - 1) / T, T, 0, stream>>>(agg2, 0.0f, N_NODES * D_OUT);
  fill_kernel<<<1, T, 0, stream>>>(pooled, 0.0f, N_GRAPHS * D_OUT);
  fill_kernel<<<1, T, 0, stream>>>(cnt, 0.0f, N_GRAPHS);
  fill_kernel<<<(N_NODES + T - 1) / T, T, 0, stream>>>(deg, 1.0f, N_NODES); // self-loop

  // 2. degree + symmetric normalization
  deg_kernel<<<(N_EDGES + T - 1) / T, T, 0, stream>>>(dst, deg);
  rsqrt_kernel<<<(N_NODES + T - 1) / T, T, 0, stream>>>(deg, dis);

  // 3. H = X @ W1 (WMMA): 6250 M-tiles, 8 waves/block cover N=128
  gemm1_wmma<<<N_NODES / 16, 256, 0, stream>>>(x, W1, H);

  // 4. layer-1 scatter: one wave per (edge + self-loop)
  {
    const int total = N_EDGES + N_NODES;
    const int wavesPerBlock = T / 32;
    edge_agg1<<<(total + wavesPerBlock - 1) / wavesPerBlock, T, 0, stream>>>(
        src, dst, dis, H, agg1);
  }

  // 5. h2 = LeakyReLU(agg1 + b1) @ W2
  h2_kernel<<<(N_NODES + T - 1) / T, T, 0, stream>>>(agg1, b1, W2, h2);

  // 6. layer-2 scatter (16B messages)
  {
    const int total = N_EDGES + N_NODES;
    edge_agg2<<<(total + T - 1) / T, T, 0, stream>>>(src, dst, dis, h2, agg2);
  }

  // 7. mean pool + softmax
  pool_kernel<<<(N_NODES + T - 1) / T, T, 0, stream>>>(bat, agg2, pooled, cnt);
  final_kernel<<<1, N_GRAPHS, 0, stream>>>(pooled, cnt, b2, out);
}